// SIMPLESampler_32478542693127
// MI455X (gfx1250) — compile-verified
//
#include <hip/hip_runtime.h>
#include <hip/hip_bf16.h>
#include <stdint.h>

// SIMPLE k-subset sampler, shape-specialized: scores (nnodes, 32, 2), k=4, S=2.
// Linear-space tree DP per row (one row per lane, wave32), async-LDS staged I/O.

struct T5 { float c1, c2, c3, c4; };   // subtree count-poly coefficients; c0 == 1 implicit
struct G5 { float g0, g1, g2, g3, g4; };

__device__ __forceinline__ T5 merge5(const T5 a, const T5 b) {
  // c[j] = sum_i a[i]*b[j-i], truncated at degree 4 (K=5), a0=b0=1
  T5 c;
  c.c1 = a.c1 + b.c1;
  c.c2 = a.c2 + a.c1 * b.c1 + b.c2;
  c.c3 = a.c3 + a.c2 * b.c1 + a.c1 * b.c2 + b.c3;
  c.c4 = a.c4 + a.c3 * b.c1 + a.c2 * b.c2 + a.c1 * b.c3 + b.c4;
  return c;
}

__device__ __forceinline__ float selT(const T5 t, int i) {
  // t[i] with t[0]==1, out-of-range -> 0 (cndmask chain; no runtime array indexing)
  float r = 0.f;
  r = (i == 0) ? 1.f  : r;
  r = (i == 1) ? t.c1 : r;
  r = (i == 2) ? t.c2 : r;
  r = (i == 3) ? t.c3 : r;
  r = (i == 4) ? t.c4 : r;
  return r;
}

__device__ __forceinline__ G5 corr5(const G5 g, const T5 s) {
  // child gradient: gA[i] = sum_{j>=i} g[j] * sib[j-i]   (sib[0]==1)
  G5 r;
  r.g0 = g.g0 + g.g1 * s.c1 + g.g2 * s.c2 + g.g3 * s.c3 + g.g4 * s.c4;
  r.g1 = g.g1 + g.g2 * s.c1 + g.g3 * s.c2 + g.g4 * s.c3;
  r.g2 = g.g2 + g.g3 * s.c1 + g.g4 * s.c2;
  r.g3 = g.g3 + g.g4 * s.c1;
  r.g4 = g.g4;
  return r;
}

__device__ __forceinline__ float urand(unsigned r, unsigned node, unsigned s) {
  unsigned a = r * 0x9E3779B9u + node * 0x01000193u + s * 0x85EBCA6Bu + 0x7F4A7C15u;
  a ^= a >> 16; a *= 0x7FEB352Du;
  a ^= a >> 15; a *= 0x846CA68Bu;
  a ^= a >> 16;
  return (float)(a >> 8) * 0x1p-24f;   // [0,1)
}

__device__ __forceinline__ int split5(const T5 A, const T5 B, int m, float u) {
  // categorical over j: P(j) ~ A[j]*B[m-j]; inverse-CDF draw
  float q0 =        selT(B, m);
  float q1 = A.c1 * selT(B, m - 1);
  float q2 = A.c2 * selT(B, m - 2);
  float q3 = A.c3 * selT(B, m - 3);
  float q4 = A.c4 * selT(B, m - 4);
  float p1 = q0 + q1, p2 = p1 + q2, p3 = p2 + q3, p4 = p3 + q4;
  float x = u * p4;
  return (int)(x > q0) + (int)(x > p1) + (int)(x > p2) + (int)(x > p3);
}

#define TILE_FLOATS 8192   // 128 nodes * 64 floats = 32 KB per tile

__global__ __launch_bounds__(256)
void simple_sampler_kernel(const float* __restrict__ scores,
                           float* __restrict__ out,
                           int nnodes) {
  __shared__ float tile_in[TILE_FLOATS];   // input thetas, later reused for marginals
  __shared__ float tile_s0[TILE_FLOATS];   // sample 0 staging
  __shared__ float tile_s1[TILE_FLOATS];   // sample 1 staging

  const unsigned tid    = threadIdx.x;
  const unsigned n0     = blockIdx.x * 128u;
  const unsigned vnodes = ((unsigned)nnodes - n0 < 128u) ? ((unsigned)nnodes - n0) : 128u;
  const unsigned vbytes = vnodes * 256u;   // valid bytes in this block's tile

  const unsigned lds_in = (unsigned)(uintptr_t)(void*)tile_in;
  const unsigned lds_s0 = (unsigned)(uintptr_t)(void*)tile_s0;
  const unsigned lds_s1 = (unsigned)(uintptr_t)(void*)tile_s1;

  // ---- Phase 1: async copy 32KB contiguous input tile into LDS (CDNA5 ASYNCcnt path)
  {
    const unsigned long long gbase =
        (unsigned long long)(uintptr_t)scores + (unsigned long long)n0 * 256ull;
#pragma unroll
    for (int i = 0; i < 8; ++i) {
      unsigned rel = tid * 16u + (unsigned)i * 4096u;
      rel = (rel > vbytes - 16u) ? (vbytes - 16u) : rel;   // tail: clamped duplicate copy
      unsigned l = lds_in + rel;
      unsigned long long g = gbase + rel;
      asm volatile("global_load_async_to_lds_b128 %0, %1, off"
                   :: "v"(l), "v"(g) : "memory");
    }
  }
  asm volatile("s_wait_asynccnt 0" ::: "memory");
  __syncthreads();

  // ---- Phase 2: per-lane DP. Row r = 2*n + e; this lane's theta at stride 2 in LDS.
  const unsigned fib = (tid >> 1) * 64u + (tid & 1u);
  const unsigned row = n0 * 2u + tid;

  float w[32];
#pragma unroll
  for (int c = 0; c < 32; ++c) w[c] = __expf(tile_in[fib + 2u * (unsigned)c]);

  // Up pass: elementary-symmetric-polynomial tables, degree-clipped at 4.
  T5 t1[16];
#pragma unroll
  for (int i = 0; i < 16; ++i) {
    float a = w[2 * i], b = w[2 * i + 1];
    t1[i].c1 = a + b; t1[i].c2 = a * b; t1[i].c3 = 0.f; t1[i].c4 = 0.f;
  }
  T5 t2[8];
#pragma unroll
  for (int i = 0; i < 8; ++i) t2[i] = merge5(t1[2 * i], t1[2 * i + 1]);
  T5 t3[4];
#pragma unroll
  for (int i = 0; i < 4; ++i) t3[i] = merge5(t2[2 * i], t2[2 * i + 1]);
  T5 t4[2];
#pragma unroll
  for (int i = 0; i < 2; ++i) t4[i] = merge5(t3[2 * i], t3[2 * i + 1]);

  const float Z = t4[0].c4 + t4[0].c3 * t4[1].c1 + t4[0].c2 * t4[1].c2
                + t4[0].c1 * t4[1].c3 + t4[1].c4;      // e_4 of all 32 weights
  const float invZ = __builtin_amdgcn_rcpf(Z);

  // Root split for both samples (root holds exactly k=4 selections).
  int c4cnt[2][2];
#pragma unroll
  for (int s = 0; s < 2; ++s) {
    int jl = split5(t4[0], t4[1], 4, urand(row, 0u, (unsigned)s));
    c4cnt[0][s] = jl; c4cnt[1][s] = 4 - jl;
  }

  // Fused top-down DFS: gradients (marginals) + conditional sampling, fully unrolled.
#pragma unroll
  for (int a = 0; a < 2; ++a) {
    const T5 sib4 = t4[1 - a];
    G5 g4; g4.g0 = sib4.c4; g4.g1 = sib4.c3; g4.g2 = sib4.c2; g4.g3 = sib4.c1; g4.g4 = 1.f;
    int c3cnt[2][2];
#pragma unroll
    for (int s = 0; s < 2; ++s) {
      int m  = c4cnt[a][s];
      int jl = split5(t3[2 * a], t3[2 * a + 1], m, urand(row, 1u + (unsigned)a, (unsigned)s));
      c3cnt[0][s] = jl; c3cnt[1][s] = m - jl;
    }
#pragma unroll
    for (int b = 0; b < 2; ++b) {
      const int n3 = 2 * a + b;
      G5 g3 = corr5(g4, t3[2 * a + (1 - b)]);
      int c2cnt[2][2];
#pragma unroll
      for (int s = 0; s < 2; ++s) {
        int m  = c3cnt[b][s];
        int jl = split5(t2[2 * n3], t2[2 * n3 + 1], m, urand(row, 3u + (unsigned)n3, (unsigned)s));
        c2cnt[0][s] = jl; c2cnt[1][s] = m - jl;
      }
#pragma unroll
      for (int cx = 0; cx < 2; ++cx) {
        const int n2 = 2 * n3 + cx;
        G5 g2 = corr5(g3, t2[2 * n3 + (1 - cx)]);
        int c1cnt[2][2];
#pragma unroll
        for (int s = 0; s < 2; ++s) {
          int m  = c2cnt[cx][s];
          int jl = split5(t1[2 * n2], t1[2 * n2 + 1], m, urand(row, 7u + (unsigned)n2, (unsigned)s));
          c1cnt[0][s] = jl; c1cnt[1][s] = m - jl;
        }
#pragma unroll
        for (int d = 0; d < 2; ++d) {
          const int n1 = 2 * n2 + d;
          G5 g1 = corr5(g2, t1[2 * n2 + (1 - d)]);
          const float wa = w[2 * n1], wb = w[2 * n1 + 1];
          const float mA = wa * (g1.g1 + g1.g2 * wb) * invZ;   // exact marginal
          const float mB = wb * (g1.g1 + g1.g2 * wa) * invZ;
          const unsigned ia = fib + 4u * (unsigned)n1;         // lds idx of leaf 2*n1
          const unsigned ib = ia + 2u;
          tile_in[ia] = mA; tile_in[ib] = mB;                  // own slots only: no race
#pragma unroll
          for (int s = 0; s < 2; ++s) {
            int m = c1cnt[d][s];                               // 0..2
            float q0 =      ((m == 0) ? 1.f : ((m == 1) ? wb : 0.f));
            float q1 = wa * ((m == 1) ? 1.f : ((m == 2) ? wb : 0.f));
            float u  = urand(row, 15u + (unsigned)n1, (unsigned)s);
            int jl = (u * (q0 + q1) > q0) ? 1 : 0;
            int jr = m - jl;
            float sa = ((float)jl - mA) + mA;                  // straight-through fwd value
            float sb = ((float)jr - mB) + mB;
            float* t = s ? tile_s1 : tile_s0;
            t[ia] = sa; t[ib] = sb;
          }
        }
      }
    }
  }

  // ---- Phase 3: async drain of the three 32KB tiles to the packed output.
  __syncthreads();
  {
    const unsigned long long ob  = (unsigned long long)(uintptr_t)out;
    const unsigned long long nb  = (unsigned long long)(unsigned)nnodes * 256ull; // one S-slab
    const unsigned long long s0b = ob + (unsigned long long)n0 * 256ull;
    const unsigned long long s1b = s0b + nb;            // samples[1]
    const unsigned long long mgb = s0b + 2ull * nb;     // marginals after samples
#pragma unroll
    for (int i = 0; i < 8; ++i) {
      unsigned rel = tid * 16u + (unsigned)i * 4096u;
      rel = (rel > vbytes - 16u) ? (vbytes - 16u) : rel;
      unsigned l0 = lds_s0 + rel, l1 = lds_s1 + rel, lm = lds_in + rel;
      unsigned long long a0 = s0b + rel, a1 = s1b + rel, am = mgb + rel;
      asm volatile("global_store_async_from_lds_b128 %0, %1, off" :: "v"(a0), "v"(l0) : "memory");
      asm volatile("global_store_async_from_lds_b128 %0, %1, off" :: "v"(a1), "v"(l1) : "memory");
      asm volatile("global_store_async_from_lds_b128 %0, %1, off" :: "v"(am), "v"(lm) : "memory");
    }
  }
  asm volatile("s_wait_asynccnt 0" ::: "memory");
}

extern "C" void kernel_launch(void* const* d_in, const int* in_sizes, int n_in,
                              void* d_out, int out_size, void* d_ws, size_t ws_size,
                              hipStream_t stream) {
  (void)n_in; (void)out_size; (void)d_ws; (void)ws_size;
  const float* scores = (const float*)d_in[0];
  float* out = (float*)d_out;
  const int nnodes = in_sizes[0] / 64;          // (nnodes, 32, 2) fp32
  const int rows   = nnodes * 2;
  const int blocks = (rows + 255) / 256;        // 128 nodes per block
  simple_sampler_kernel<<<blocks, 256, 0, stream>>>(scores, out, nnodes);
}